// LabelGraphRefiner_56719338111441
// MI455X (gfx1250) — compile-verified
//
#include <hip/hip_runtime.h>

#define F_IN 384
#define H1   128
#define H2   64

typedef __attribute__((ext_vector_type(2))) float v2f;
typedef __attribute__((ext_vector_type(8))) float v8f;

// ---------------- degree / norm ----------------

__global__ void deg_init_kernel(float* deg, int n) {
  int i = blockIdx.x * blockDim.x + threadIdx.x;
  if (i < n) deg[i] = 1.0f;  // self-loop contributes 1
}

__global__ void deg_count_kernel(const int* __restrict__ dst, float* deg, int e) {
  int i = blockIdx.x * blockDim.x + threadIdx.x;
  if (i < e) atomicAdd(&deg[dst[i]], 1.0f);
}

__global__ void deg_rsqrt_kernel(float* deg, int n) {
  int i = blockIdx.x * blockDim.x + threadIdx.x;
  if (i < n) deg[i] = rsqrtf(deg[i]);  // deg >= 1 always (self-loop)
}

// ---------------- fp32 WMMA GEMM: C[rows x NC] = A[rows x K] * B[K x NC] ----------------
// One wave per 16x16 output tile; V_WMMA_F32_16X16X4_F32 over K in steps of 4.

template<int K, int NC>
__global__ void gemm_wmma_f32_kernel(const float* __restrict__ A,
                                     const float* __restrict__ B,
                                     float* __restrict__ C,
                                     int rowTiles) {
  constexpr int colTiles = NC / 16;
  int wave = blockIdx.x * (blockDim.x >> 5) + (threadIdx.x >> 5);
  int tr = wave / colTiles;
  int tc = wave % colTiles;
  if (tr >= rowTiles) return;          // wave-uniform; never taken with exact grids
  int lane = threadIdx.x & 31;
  int half = lane >> 4;                // 0: K+0/K+1 ; 1: K+2/K+3
  int lid  = lane & 15;

  const float* Ap = A + (size_t)(tr * 16 + lid) * K;   // row M = lid
  const float* Bp = B + tc * 16 + lid;                 // col N = lid

  v8f acc = {};
  for (int k = 0; k < K; k += 4) {
    int ka = k + half * 2;
    v2f a, b;
    a.x = Ap[ka];
    a.y = Ap[ka + 1];
    b.x = Bp[(size_t)ka * NC];
    b.y = Bp[(size_t)(ka + 1) * NC];
    acc = __builtin_amdgcn_wmma_f32_16x16x4_f32(false, a, false, b,
                                                (short)0, acc, false, false);
  }

  // C/D layout: VGPR j -> row j + half*8, col = lid
  float* Cp = C + (size_t)(tr * 16 + half * 8) * NC + tc * 16 + lid;
#pragma unroll
  for (int j = 0; j < 8; ++j) Cp[(size_t)j * NC] = acc[j];
}

// ---------------- self-loop init / bias / relu ----------------

template<int F>
__global__ void self_msg_kernel(const float* __restrict__ dis,
                                const float* __restrict__ t,
                                float* __restrict__ out, int n) {
  long idx = (long)blockIdx.x * blockDim.x + threadIdx.x;
  if (idx >= (long)n * F) return;
  int i = (int)(idx / F);
  float d = dis[i];
  out[idx] = d * d * t[idx];
}

template<int F>
__global__ void self_msg_bias_kernel(const float* __restrict__ dis,
                                     const float* __restrict__ t,
                                     const float* __restrict__ bias,
                                     float* __restrict__ out, int n) {
  long idx = (long)blockIdx.x * blockDim.x + threadIdx.x;
  if (idx >= (long)n * F) return;
  int i = (int)(idx / F);
  int f = (int)(idx % F);
  float d = dis[i];
  out[idx] = d * d * t[idx] + bias[f];
}

template<int F>
__global__ void bias_relu_kernel(float* __restrict__ h,
                                 const float* __restrict__ bias, int n) {
  long idx = (long)blockIdx.x * blockDim.x + threadIdx.x;
  if (idx >= (long)n * F) return;
  int f = (int)(idx % F);
  h[idx] = fmaxf(h[idx] + bias[f], 0.0f);
}

// ---------------- edge gather/scale/scatter (one wave32 per edge) ----------------

__global__ void edge_scatter128_kernel(const int* __restrict__ src,
                                       const int* __restrict__ dst,
                                       const float* __restrict__ dis,
                                       const float* __restrict__ msg,
                                       float* out, int numEdges) {
  int gid = blockIdx.x * blockDim.x + threadIdx.x;
  int e = gid >> 5;
  if (e >= numEdges) return;
  int lane = gid & 31;
  int s = src[e], d = dst[e];
  float c = dis[s] * dis[d];
  const float4 m = *reinterpret_cast<const float4*>(msg + (size_t)s * 128 + lane * 4);
  float* op = out + (size_t)d * 128 + lane * 4;
  atomicAdd(op + 0, c * m.x);
  atomicAdd(op + 1, c * m.y);
  atomicAdd(op + 2, c * m.z);
  atomicAdd(op + 3, c * m.w);
}

__global__ void edge_scatter64_kernel(const int* __restrict__ src,
                                      const int* __restrict__ dst,
                                      const float* __restrict__ dis,
                                      const float* __restrict__ msg,
                                      float* out, int numEdges) {
  int gid = blockIdx.x * blockDim.x + threadIdx.x;
  int e = gid >> 5;
  if (e >= numEdges) return;
  int lane = gid & 31;
  int s = src[e], d = dst[e];
  float c = dis[s] * dis[d];
  const float2 m = *reinterpret_cast<const float2*>(msg + (size_t)s * 64 + lane * 2);
  float* op = out + (size_t)d * 64 + lane * 2;
  atomicAdd(op + 0, c * m.x);
  atomicAdd(op + 1, c * m.y);
}

// ---------------- launch ----------------

extern "C" void kernel_launch(void* const* d_in, const int* in_sizes, int n_in,
                              void* d_out, int out_size, void* d_ws, size_t ws_size,
                              hipStream_t stream) {
  (void)n_in; (void)out_size; (void)ws_size;

  const float* x  = (const float*)d_in[0];
  const float* W1 = (const float*)d_in[1];
  const float* b1 = (const float*)d_in[2];
  const float* W2 = (const float*)d_in[3];
  const float* b2 = (const float*)d_in[4];
  const int*   ei = (const int*)d_in[5];

  const int N = in_sizes[0] / F_IN;   // 100000
  const int E = in_sizes[5] / 2;      // 3200000
  const int* srcIdx = ei;
  const int* dstIdx = ei + E;

  float* ws  = (float*)d_ws;
  float* dis = ws;
  size_t disPad = ((size_t)N + 255) & ~(size_t)255;
  float* t   = ws + disPad;                 // N x 128 (reused for N x 64)
  float* agg = t + (size_t)N * H1;          // N x 128
  float* out = (float*)d_out;               // N x 64

  const dim3 blk(256);

  // norm
  deg_init_kernel<<<(N + 255) / 256, blk, 0, stream>>>(dis, N);
  deg_count_kernel<<<(E + 255) / 256, blk, 0, stream>>>(dstIdx, dis, E);
  deg_rsqrt_kernel<<<(N + 255) / 256, blk, 0, stream>>>(dis, N);

  const int rowTiles = N / 16;              // 6250

  // layer 1
  int waves1 = rowTiles * (H1 / 16);        // 50000 -> 6250 blocks of 8 waves
  gemm_wmma_f32_kernel<F_IN, H1><<<(waves1 + 7) / 8, blk, 0, stream>>>(x, W1, t, rowTiles);

  long tot1 = (long)N * H1;
  self_msg_kernel<H1><<<(int)((tot1 + 255) / 256), blk, 0, stream>>>(dis, t, agg, N);
  long ethreads = (long)E * 32;
  edge_scatter128_kernel<<<(int)((ethreads + 255) / 256), blk, 0, stream>>>(
      srcIdx, dstIdx, dis, t, agg, E);
  bias_relu_kernel<H1><<<(int)((tot1 + 255) / 256), blk, 0, stream>>>(agg, b1, N);

  // layer 2
  int waves2 = rowTiles * (H2 / 16);        // 25000 -> 3125 blocks
  gemm_wmma_f32_kernel<H1, H2><<<(waves2 + 7) / 8, blk, 0, stream>>>(agg, W2, t, rowTiles);

  long tot2 = (long)N * H2;
  self_msg_bias_kernel<H2><<<(int)((tot2 + 255) / 256), blk, 0, stream>>>(dis, t, b2, out, N);
  edge_scatter64_kernel<<<(int)((ethreads + 255) / 256), blk, 0, stream>>>(
      srcIdx, dstIdx, dis, t, out, E);
}